// AdvancedClinicalSafetyLoss_84146999263814
// MI455X (gfx1250) — compile-verified
//
#include <hip/hip_runtime.h>

#define NTHR 256
#define WAVES_PER_BLK (NTHR / 32)

typedef float v2f __attribute__((ext_vector_type(2)));
typedef float v8f __attribute__((ext_vector_type(8)));

// Full 32-lane sum using V_WMMA_F32_16X16X4_F32.
// A layout (16x4 f32): lanes 0-15 supply A[m][0..1], lanes 16-31 supply A[m][2..3].
// With A = (v, 0) and B = ones: D[m][n] = v_m + v_{m+16} for all n.
// Lane n holds D[0..7][n] (lanes 16-31: D[8..15][n]) in 8 VGPRs; summing them
// gives each half-wave half the total, fixed with one shfl_xor(16).
__device__ __forceinline__ float wave_sum_wmma(float v) {
    v2f a; a[0] = v;    a[1] = 0.0f;
    v2f b; b[0] = 1.0f; b[1] = 1.0f;
    v8f c = {};
    v8f d = __builtin_amdgcn_wmma_f32_16x16x4_f32(
        /*neg_a=*/false, a, /*neg_b=*/false, b,
        /*c_mod=*/(short)0, c, /*reuse_a=*/false, /*reuse_b=*/false);
    float s = ((d[0] + d[1]) + (d[2] + d[3])) + ((d[4] + d[5]) + (d[6] + d[7]));
    s += __shfl_xor(s, 16, 32);
    return s;
}

__device__ __forceinline__ void proc_sample(
    float o0, float o1, float o2, int t,
    const float* sCW, const float* sPen,
    float& a_wce, float& a_w, float& a_f, float& a_p, float& a_nc, float& a_ms)
{
    // log-sum-exp (f32)
    float m  = fmaxf(fmaxf(o0, o1), o2);       // v_max3
    float e0 = __expf(o0 - m);
    float e1 = __expf(o1 - m);
    float e2 = __expf(o2 - m);
    float s  = e0 + e1 + e2;
    float ot = (t == 0) ? o0 : ((t == 1) ? o1 : o2);
    float et = (t == 0) ? e0 : ((t == 1) ? e1 : e2);
    float ce = __logf(s) + (m - ot);           // ce_raw >= 0

    float w = sCW[t];
    a_wce = fmaf(w, ce, a_wce);
    a_w  += w;

    // pt = exp(-ce) = et / s; fast v_rcp_f32 (~1 ulp) instead of IEEE divide
    float pt = et * __builtin_amdgcn_rcpf(s);
    float om = 1.0f - pt;
    a_f = fmaf(om * om, ce, a_f);              // ALPHA applied at finalize

    // argmax, first-occurrence tie-break (matches jnp.argmax)
    int pred = 0; float best = o0;
    if (o1 > best) { best = o1; pred = 1; }
    if (o2 > best) { pred = 2; }

    a_p += sPen[t * 3 + pred];

    bool crit = (t == 2);
    a_nc += crit ? 1.0f : 0.0f;
    a_ms += (crit && pred != 2) ? 1.0f : 0.0f;
}

__global__ __launch_bounds__(NTHR) void loss_partials_kernel(
    const float* __restrict__ outputs, const int* __restrict__ targets,
    const float* __restrict__ cw, const float* __restrict__ pm,
    double* __restrict__ ws, int quads, int n_total, int nblk)
{
    __shared__ float sCW[3];
    __shared__ float sPen[9];
    __shared__ float sWave[WAVES_PER_BLK][6];

    const int tid = threadIdx.x;
    if (tid < 9) sPen[tid] = pm[tid];
    if (tid < 3) sCW[tid]  = cw[tid];
    __syncthreads();

    float a_wce = 0.0f, a_w = 0.0f, a_f = 0.0f, a_p = 0.0f, a_nc = 0.0f, a_ms = 0.0f;

    const float4* __restrict__ o4 = (const float4*)outputs;
    const int4*   __restrict__ t4 = (const int4*)targets;

    const int g      = blockIdx.x * NTHR + tid;
    const int stride = nblk * NTHR;

    #pragma unroll 2
    for (int j = g; j < quads; j += stride) {
        float4 f0 = o4[3 * j + 0];
        float4 f1 = o4[3 * j + 1];
        float4 f2 = o4[3 * j + 2];
        int4   tq = t4[j];
        proc_sample(f0.x, f0.y, f0.z, tq.x, sCW, sPen, a_wce, a_w, a_f, a_p, a_nc, a_ms);
        proc_sample(f0.w, f1.x, f1.y, tq.y, sCW, sPen, a_wce, a_w, a_f, a_p, a_nc, a_ms);
        proc_sample(f1.z, f1.w, f2.x, tq.z, sCW, sPen, a_wce, a_w, a_f, a_p, a_nc, a_ms);
        proc_sample(f2.y, f2.z, f2.w, tq.w, sCW, sPen, a_wce, a_w, a_f, a_p, a_nc, a_ms);
    }

    // Scalar tail if B % 4 != 0 (not hit for B = 8388608; re-converges before WMMA)
    if (g == 0) {
        for (int i = quads * 4; i < n_total; ++i) {
            proc_sample(outputs[3 * i], outputs[3 * i + 1], outputs[3 * i + 2],
                        targets[i], sCW, sPen, a_wce, a_w, a_f, a_p, a_nc, a_ms);
        }
    }

    // Wave-level reduction via WMMA (EXEC all-1s here: straight-line, full block)
    float r0 = wave_sum_wmma(a_wce);
    float r1 = wave_sum_wmma(a_w);
    float r2 = wave_sum_wmma(a_f);
    float r3 = wave_sum_wmma(a_p);
    float r4 = wave_sum_wmma(a_nc);
    float r5 = wave_sum_wmma(a_ms);

    const int lane = tid & 31;
    const int wave = tid >> 5;
    if (lane == 0) {
        sWave[wave][0] = r0; sWave[wave][1] = r1; sWave[wave][2] = r2;
        sWave[wave][3] = r3; sWave[wave][4] = r4; sWave[wave][5] = r5;
    }
    __syncthreads();

    if (tid < 6) {
        double s = 0.0;
        #pragma unroll
        for (int w2 = 0; w2 < WAVES_PER_BLK; ++w2) s += (double)sWave[w2][tid];
        ws[tid * nblk + blockIdx.x] = s;   // SoA: coalesced in finalize
    }
}

__global__ __launch_bounds__(NTHR) void loss_finalize_kernel(
    const double* __restrict__ ws, float* __restrict__ out, int nblk, double nsamp)
{
    __shared__ double lds[6][NTHR];
    const int tid = threadIdx.x;

    double acc[6] = {0.0, 0.0, 0.0, 0.0, 0.0, 0.0};
    for (int i = tid; i < nblk; i += NTHR) {
        #pragma unroll
        for (int c = 0; c < 6; ++c) acc[c] += ws[c * nblk + i];
    }
    #pragma unroll
    for (int c = 0; c < 6; ++c) lds[c][tid] = acc[c];
    __syncthreads();

    for (int s = NTHR / 2; s > 0; s >>= 1) {
        if (tid < s) {
            #pragma unroll
            for (int c = 0; c < 6; ++c) lds[c][tid] += lds[c][tid + s];
        }
        __syncthreads();
    }

    if (tid == 0) {
        double sum_wce = lds[0][0];
        double sum_w   = lds[1][0];
        double sum_f   = lds[2][0];
        double sum_p   = lds[3][0];
        double ncrit   = lds[4][0];
        double miss    = lds[5][0];

        double ce_loss = sum_wce / sum_w;
        double focal   = 0.25 * sum_f / nsamp;
        double pen     = sum_p / nsamp;
        double crit    = (ncrit > 0.0) ? (miss / fmax(ncrit, 1.0)) * 50.0 : 0.0;

        out[0] = (float)(ce_loss + 0.3 * focal + 0.4 * pen + 0.6 * crit);
    }
}

extern "C" void kernel_launch(void* const* d_in, const int* in_sizes, int n_in,
                              void* d_out, int out_size, void* d_ws, size_t ws_size,
                              hipStream_t stream) {
    const float* outputs = (const float*)d_in[0];   // [B,3] f32
    const int*   targets = (const int*)d_in[1];     // [B] int32
    const float* cw      = (const float*)d_in[2];   // [3] f32
    const float* pm      = (const float*)d_in[3];   // [3,3] f32
    double* ws  = (double*)d_ws;
    float*  out = (float*)d_out;

    const int n     = in_sizes[1];   // B
    const int quads = n >> 2;

    int nblk = 2048;
    while (nblk > 1 && (size_t)nblk * 6 * sizeof(double) > ws_size) nblk >>= 1;

    loss_partials_kernel<<<nblk, NTHR, 0, stream>>>(outputs, targets, cw, pm,
                                                    ws, quads, n, nblk);
    loss_finalize_kernel<<<1, NTHR, 0, stream>>>(ws, out, nblk, (double)n);
}